// SSM_1056561955004
// MI455X (gfx1250) — compile-verified
//
#include <hip/hip_runtime.h>
#include <hip/hip_bf16.h>
#include <stdint.h>

typedef __attribute__((ext_vector_type(2))) float v2f;
typedef __attribute__((ext_vector_type(8))) float v8f;

// Issue an async global->LDS copy of 16 contiguous bytes (per lane).
// ISA: GLOBAL_LOAD_ASYNC_TO_LDS_B128 (opcode 98), tracked by ASYNCcnt.
// LDS destination byte offset = low 32 bits of the flat shared address
// (LDS aperture tag occupies addr[63:32] on CDNA5).
__device__ __forceinline__ void async_copy_b128(const void* lds_dst, const void* gsrc)
{
    uint32_t l = (uint32_t)(uintptr_t)lds_dst;
    asm volatile("global_load_async_to_lds_b128 %0, %1, off"
                 :: "v"(l), "v"(gsrc) : "memory");
}

// ---------------------------------------------------------------------------
// LDS-staged, double-buffered WMMA f32 GEMM: C = A @ B, all row-major NxNxN
// (N multiple of 128). 256 threads = 8 waves; 128x128 block tile; 32x64 per
// wave = 2x4 WMMA 16x16 tiles; K staged in chunks of 16 via async-to-LDS.
// ---------------------------------------------------------------------------
__global__ __launch_bounds__(256)
void gemm_f32_wmma_lds(const float* A, const float* B, float* __restrict__ C, int N)
{
    __shared__ float As[2][128][20];   // [row][k], pad 20: aligned + conflict-free
    __shared__ float Bs[2][16][132];   // [k][col], pad 132

    const int t    = threadIdx.x;
    const int lane = t & 31;
    const int wv   = t >> 5;
    const int wr   = wv >> 1;          // 0..3
    const int wc   = wv & 1;           // 0..1
    const int m16  = lane & 15;
    const int kh   = lane >> 4;        // 0 or 1

    const int rowBase = blockIdx.y * 128;
    const int colBase = blockIdx.x * 128;

    // copy mapping: A tile 128x16 = 512 b128-chunks, B tile 16x128 = 512 chunks
    const int ar0 = t >> 2;            // A chunk row (second chunk: +64)
    const int ak0 = (t & 3) * 4;       // A chunk k0: 0,4,8,12
    const int bk0 = t >> 5;            // B chunk k (second chunk: +8)
    const int bc0 = (t & 31) * 4;      // B chunk col0 (coalesced)

    v8f acc[2][4] = {};

    const int nStages = N / 16;

    // prologue: stage 0 into buffer 0
    async_copy_b128(&As[0][ar0][ak0],      A + (size_t)(rowBase + ar0) * N + ak0);
    async_copy_b128(&As[0][ar0 + 64][ak0], A + (size_t)(rowBase + ar0 + 64) * N + ak0);
    async_copy_b128(&Bs[0][bk0][bc0],      B + (size_t)bk0 * N + colBase + bc0);
    async_copy_b128(&Bs[0][bk0 + 8][bc0],  B + (size_t)(bk0 + 8) * N + colBase + bc0);

    for (int s = 0; s < nStages; ++s) {
        const int buf = s & 1;

        if (s + 1 < nStages) {
            const int kk = (s + 1) * 16;
            const int nb = buf ^ 1;
            async_copy_b128(&As[nb][ar0][ak0],      A + (size_t)(rowBase + ar0) * N + kk + ak0);
            async_copy_b128(&As[nb][ar0 + 64][ak0], A + (size_t)(rowBase + ar0 + 64) * N + kk + ak0);
            async_copy_b128(&Bs[nb][bk0][bc0],      B + (size_t)(kk + bk0) * N + colBase + bc0);
            async_copy_b128(&Bs[nb][bk0 + 8][bc0],  B + (size_t)(kk + bk0 + 8) * N + colBase + bc0);
            // previous stage (first 4 of the 8 outstanding) must be complete
            asm volatile("s_wait_asynccnt 4" ::: "memory");
        } else {
            asm volatile("s_wait_asynccnt 0" ::: "memory");
        }
        __syncthreads();   // tile visible to all waves

        #pragma unroll
        for (int ks = 0; ks < 4; ++ks) {
            const int ka = ks * 4 + 2 * kh;
            v2f a[2], b[4];
            #pragma unroll
            for (int rt = 0; rt < 2; ++rt) {
                const float* p = &As[buf][wr * 32 + rt * 16 + m16][ka];
                a[rt][0] = p[0];            // A[row][ka]   (one ds_load_b64)
                a[rt][1] = p[1];            // A[row][ka+1]
            }
            #pragma unroll
            for (int ct = 0; ct < 4; ++ct) {
                const float* p = &Bs[buf][ka][wc * 64 + ct * 16 + m16];
                b[ct][0] = p[0];            // B[ka][col]
                b[ct][1] = p[132];          // B[ka+1][col]
            }
            #pragma unroll
            for (int rt = 0; rt < 2; ++rt)
                #pragma unroll
                for (int ct = 0; ct < 4; ++ct)
                    acc[rt][ct] = __builtin_amdgcn_wmma_f32_16x16x4_f32(
                        false, a[rt], false, b[ct], (short)0, acc[rt][ct], false, false);
        }
        __syncthreads();   // done reading before next stage overwrites this buffer
    }

    #pragma unroll
    for (int rt = 0; rt < 2; ++rt) {
        #pragma unroll
        for (int ct = 0; ct < 4; ++ct) {
            const int col   = colBase + wc * 64 + ct * 16 + m16;
            const int rbase = rowBase + wr * 32 + rt * 16 + kh * 8;
            #pragma unroll
            for (int v = 0; v < 8; ++v)
                C[(size_t)(rbase + v) * N + col] = acc[rt][ct][v];
        }
    }
}

// ---------------------------------------------------------------------------
// General-stride WMMA f32 GEMM (direct from global) — used for the skinny
// column-major Krylov panels (Nc=32), which are L2-resident and memory-light.
// ---------------------------------------------------------------------------
__global__ __launch_bounds__(256)
void gemm_f32_wmma(const float* __restrict__ A, int Ars, int Acs,
                   const float* __restrict__ B, int Brs, int Bcs,
                   float* __restrict__ C, int Crs, int Ccs,
                   int M, int Nc, int Kd)
{
    const int lane = threadIdx.x & 31;
    const int wv   = threadIdx.x >> 5;
    const int wr   = wv >> 1;
    const int wc   = wv & 1;
    const int m16  = lane & 15;
    const int kh   = lane >> 4;

    const int row0 = blockIdx.y * 128 + wr * 32;
    const int col0 = blockIdx.x * 128 + wc * 64;

    int arow[2];
    int bcol[4];
    #pragma unroll
    for (int rt = 0; rt < 2; ++rt) arow[rt] = row0 + rt * 16 + m16;
    #pragma unroll
    for (int ct = 0; ct < 4; ++ct) {
        int c = col0 + ct * 16 + m16;
        bcol[ct] = (c < Nc) ? c : (Nc - 1);
    }

    v8f acc[2][4] = {};

    #pragma unroll 4
    for (int kk = 0; kk < Kd; kk += 4) {
        const int ka = kk + 2 * kh;

        if ((kk & 127) == 0 && (kk + 128) < Kd) {
            __builtin_prefetch(A + (size_t)arow[0] * Ars + (size_t)(kk + 128) * Acs, 0, 1);
            __builtin_prefetch(B + (size_t)(kk + 128) * Brs + (size_t)bcol[0] * Bcs, 0, 1);
        }

        v2f a[2], b[4];
        #pragma unroll
        for (int rt = 0; rt < 2; ++rt) {
            const float* p = A + (size_t)arow[rt] * Ars + (size_t)ka * Acs;
            a[rt][0] = p[0];
            a[rt][1] = p[Acs];
        }
        #pragma unroll
        for (int ct = 0; ct < 4; ++ct) {
            const float* p = B + (size_t)ka * Brs + (size_t)bcol[ct] * Bcs;
            b[ct][0] = p[0];
            b[ct][1] = p[Brs];
        }
        #pragma unroll
        for (int rt = 0; rt < 2; ++rt)
            #pragma unroll
            for (int ct = 0; ct < 4; ++ct)
                acc[rt][ct] = __builtin_amdgcn_wmma_f32_16x16x4_f32(
                    false, a[rt], false, b[ct], (short)0, acc[rt][ct], false, false);
    }

    #pragma unroll
    for (int rt = 0; rt < 2; ++rt) {
        #pragma unroll
        for (int ct = 0; ct < 4; ++ct) {
            const int col = col0 + ct * 16 + m16;
            if (col < Nc) {
                const int rbase = row0 + rt * 16 + kh * 8;
                #pragma unroll
                for (int v = 0; v < 8; ++v)
                    C[(size_t)(rbase + v) * Crs + (size_t)col * Ccs] = acc[rt][ct][v];
            }
        }
    }
}

// ---------------------------------------------------------------------------
// y[row] = alpha * (M @ x)[row] + beta * z[row]   (one wave32 per row)
// ---------------------------------------------------------------------------
__global__ __launch_bounds__(256)
void matvec_f32(const float* __restrict__ M, int ld,
                const float* __restrict__ x, const float* __restrict__ z,
                float* __restrict__ y, int nrows, int ncols,
                float alpha, float beta)
{
    const int lane = threadIdx.x & 31;
    const int row  = blockIdx.x * (blockDim.x >> 5) + (threadIdx.x >> 5);
    if (row >= nrows) return;

    const float* mr = M + (size_t)row * ld;
    float s = 0.0f;
    for (int k = lane * 4; k < ncols; k += 128) {
        float4 mv = *(const float4*)(mr + k);
        float4 xv = *(const float4*)(x + k);
        s += mv.x * xv.x + mv.y * xv.y + mv.z * xv.z + mv.w * xv.w;
    }
    #pragma unroll
    for (int off = 16; off > 0; off >>= 1) s += __shfl_xor(s, off, 32);

    if (lane == 0) {
        float r = alpha * s;
        if (z != nullptr) r += beta * z[row];
        y[row] = r;
    }
}

// Ab = I + 2e*A + 2e^2*A2 + 2e^3*A3   (Neumann series of (I-eA)^{-1}(I+eA))
__global__ __launch_bounds__(256)
void build_ab(const float* __restrict__ A, const float* __restrict__ A2,
              const float* __restrict__ A3, float* __restrict__ Ab,
              int N, float e)
{
    size_t i = (size_t)blockIdx.x * blockDim.x + threadIdx.x;
    int r = (int)(i / (size_t)N);
    int c = (int)(i % (size_t)N);
    float e2 = e * e;
    float v = 2.0f * e * A[i] + 2.0f * e2 * A2[i] + 2.0f * e2 * e * A3[i];
    Ab[i] = v + ((r == c) ? 1.0f : 0.0f);
}

__global__ __launch_bounds__(256)
void scale_f32(const float* __restrict__ in, float* __restrict__ out, int n, float s)
{
    int i = blockIdx.x * blockDim.x + threadIdx.x;
    if (i < n) out[i] = s * in[i];
}

// y[k] = sum_{j=0..k} h[k-j] * u[j]   (one wave32 per output element)
__global__ __launch_bounds__(256)
void conv_causal(const float* __restrict__ h, const float* __restrict__ u,
                 float* __restrict__ y, int L)
{
    const int lane = threadIdx.x & 31;
    const int k    = blockIdx.x * (blockDim.x >> 5) + (threadIdx.x >> 5);
    if (k >= L) return;

    float s = 0.0f;
    for (int j = lane; j <= k; j += 32) s += h[k - j] * u[j];
    #pragma unroll
    for (int off = 16; off > 0; off >>= 1) s += __shfl_xor(s, off, 32);
    if (lane == 0) y[k] = s;
}

// ---------------------------------------------------------------------------
extern "C" void kernel_launch(void* const* d_in, const int* in_sizes, int n_in,
                              void* d_out, int out_size, void* d_ws, size_t ws_size,
                              hipStream_t stream)
{
    (void)n_in; (void)out_size; (void)ws_size;

    const float* A  = (const float*)d_in[0];   // (N,N)
    const float* Bv = (const float*)d_in[1];   // (N,1)
    const float* Cv = (const float*)d_in[2];   // (1,N)
    const float* u  = (const float*)d_in[3];   // (L,)
    float* y = (float*)d_out;                  // (L,1)

    const int N = in_sizes[1];                 // 4096
    const int L = in_sizes[3];                 // 2048
    const float step = 1.0f / (float)L;
    const float eps  = 0.5f * step;            // = 1/4096

    float* W0   = (float*)d_ws;                // A2 / Ab^{2,8,32}
    float* W1   = W0 + (size_t)N * N;          // A3 / Ab^{4,16}
    float* W2   = W1 + (size_t)N * N;          // Ab
    float* Kbuf = W2 + (size_t)N * N;          // Krylov, column-major N x L
    float* hbuf = Kbuf + (size_t)N * L;        // Markov params, L
    float* xa   = hbuf + L;
    float* xb   = xa + N;

    dim3 blk(256);
    dim3 gfull(N / 128, N / 128);

    // --- Neumann build of Ab (error O((3eps)^4) ~ 3e-13, below fp32 eps) ---
    gemm_f32_wmma_lds<<<gfull, blk, 0, stream>>>(A, A,  W0, N); // A^2
    gemm_f32_wmma_lds<<<gfull, blk, 0, stream>>>(A, W0, W1, N); // A^3
    build_ab<<<(unsigned)(((size_t)N * N) / 256), blk, 0, stream>>>(A, W0, W1, W2, N, eps);

    // --- Ab^32 by repeated squaring (5 GEMMs), result in W0 ---
    gemm_f32_wmma_lds<<<gfull, blk, 0, stream>>>(W2, W2, W0, N); // ^2
    gemm_f32_wmma_lds<<<gfull, blk, 0, stream>>>(W0, W0, W1, N); // ^4
    gemm_f32_wmma_lds<<<gfull, blk, 0, stream>>>(W1, W1, W0, N); // ^8
    gemm_f32_wmma_lds<<<gfull, blk, 0, stream>>>(W0, W0, W1, N); // ^16
    gemm_f32_wmma_lds<<<gfull, blk, 0, stream>>>(W1, W1, W0, N); // ^32

    // --- Bb = (I - eps A)^{-1} step B via fixed point x <- step*B + eps*A*x ---
    scale_f32<<<(N + 255) / 256, blk, 0, stream>>>(Bv, xa, N, step);
    matvec_f32<<<N / 8, blk, 0, stream>>>(A, N, xa, Bv, xb, N, N, eps, step);
    matvec_f32<<<N / 8, blk, 0, stream>>>(A, N, xb, Bv, xa, N, N, eps, step);
    matvec_f32<<<N / 8, blk, 0, stream>>>(A, N, xa, Bv, xb, N, N, eps, step);
    matvec_f32<<<N / 8, blk, 0, stream>>>(A, N, xb, Bv, Kbuf, N, N, eps, step); // K[:,0]=Bb

    // --- First 32 Krylov columns: K[:,j] = Ab @ K[:,j-1] (Ab is L2-resident) ---
    for (int j = 1; j < 32; ++j)
        matvec_f32<<<N / 8, blk, 0, stream>>>(W2, N, Kbuf + (size_t)(j - 1) * N, nullptr,
                                              Kbuf + (size_t)j * N, N, N, 1.0f, 0.0f);

    // --- Panel hops: K[:, 32i..] = Ab^32 @ K[:, 32(i-1)..]  (N x N x 32 GEMMs) ---
    dim3 gpanel(1, N / 128);
    for (int i = 1; i < L / 32; ++i)
        gemm_f32_wmma<<<gpanel, blk, 0, stream>>>(
            W0, N, 1,
            Kbuf + (size_t)(i - 1) * 32 * N, 1, N,
            Kbuf + (size_t)i * 32 * N, 1, N,
            N, 32, N);

    // --- Markov parameters h[m] = C . K[:,m] (columns contiguous, ld=N) ---
    matvec_f32<<<L / 8, blk, 0, stream>>>(Kbuf, N, Cv, nullptr, hbuf, L, N, 1.0f, 0.0f);

    // --- y[k] = sum_{j<=k} h[k-j] u[j] ---
    conv_causal<<<L / 8, blk, 0, stream>>>(hbuf, u, y, L);
}